// CorrectAndSmooth_6975026888996
// MI455X (gfx1250) — compile-verified
//
#include <hip/hip_runtime.h>
#include <cstdint>

typedef __attribute__((ext_vector_type(2))) float v2f;
typedef __attribute__((ext_vector_type(8))) float v8f;

#define NN 100000
#define EE 3200000
#define CC 64
#define NCF (NN * CC)
#define A_CORR 0.979f
#define A_SMOOTH 0.756f

// ---------------- init: zero norm, h accumulator, reduction scalars ----------------
__global__ void k_init(float* __restrict__ norm, float* __restrict__ h,
                       float* __restrict__ scal) {
  int t = blockIdx.x * 256 + threadIdx.x;
  if (t < NCF) h[t] = 0.0f;
  if (t < NN)  norm[t] = 0.0f;
  if (t < 8)   scal[t] = 0.0f;
}

// ---------------- degree accumulation (into norm buffer) ----------------
__global__ void k_deg(const int* __restrict__ dst, float* __restrict__ norm) {
  int e = blockIdx.x * 256 + threadIdx.x;
  if (e >= EE) return;
  atomicAdd(norm + dst[e], 1.0f);
}

__global__ void k_norm_fin(float* __restrict__ norm) {
  int i = blockIdx.x * 256 + threadIdx.x;
  if (i >= NN) return;
  norm[i] = 1.0f / sqrtf(fmaxf(norm[i], 1.0f));
}

// ---------------- error = mask ? y_true - y_soft : 0 ; global sums ----------------
__global__ void k_error(const float* __restrict__ ysoft, const float* __restrict__ ytrue,
                        const unsigned char* __restrict__ mask,
                        float* __restrict__ y, float* __restrict__ last,
                        float* __restrict__ scal) {
  __shared__ float s_abs, s_cnt;
  int t = blockIdx.x * 256 + threadIdx.x;
  if (threadIdx.x == 0) { s_abs = 0.0f; s_cnt = 0.0f; }
  __syncthreads();
  float e = 0.0f;
  if (t < NCF) {
    int i = t >> 6;
    bool m = mask[i] != 0;
    e = m ? (ytrue[t] - ysoft[t]) : 0.0f;
    y[t] = e;
    last[t] = (1.0f - A_CORR) * e;
    if (((t & 63) == 0) && m) atomicAdd(&s_cnt, 1.0f);
  }
  atomicAdd(&s_abs, fabsf(e));
  __syncthreads();
  if (threadIdx.x == 0) {
    atomicAdd(&scal[0], s_abs);
    atomicAdd(&scal[1], s_cnt);
  }
}

// ---------------- edge gather + scatter-add: h[dst] += norm[src]*y[src] ----------------
// One thread handles 4 channels (float4 gather, 4 f32 atomics). L2-resident working set.
__global__ void k_edge(const int* __restrict__ src, const int* __restrict__ dst,
                       const float* __restrict__ norm, const float* __restrict__ y,
                       float* __restrict__ h) {
  int t = blockIdx.x * 256 + threadIdx.x;
  if (t >= EE * 16) return;
  int e  = t >> 4;
  int c4 = (t & 15) << 2;
  int s = src[e];
  int d = dst[e];
  float ns = norm[s];
  float4 v = *(const float4*)(y + (size_t)s * CC + c4);
  float* hp = h + (size_t)d * CC + c4;
  atomicAdd(hp + 0, ns * v.x);
  atomicAdd(hp + 1, ns * v.y);
  atomicAdd(hp + 2, ns * v.z);
  atomicAdd(hp + 3, ns * v.w);
}

// ---------------- y = clip(last + alpha*norm*h, lo, hi); h reset to 0 ----------------
__global__ void k_combine(const float* __restrict__ norm, const float* __restrict__ last,
                          float* __restrict__ h, float* __restrict__ yout,
                          float alpha, float lo, float hi) {
  int t = blockIdx.x * 256 + threadIdx.x;
  if (t >= NCF) return;
  int i = t >> 6;
  float v = last[t] + alpha * norm[i] * h[t];
  v = fminf(fmaxf(v, lo), hi);
  yout[t] = v;
  h[t] = 0.0f;  // pre-zero accumulator for next layer
}

// ---------------- correct->smooth transition, WMMA row-sum reduction ----------------
// One wave (32 lanes) per 16 nodes. rowsum_i = sum_c |serr[i,c]| computed as
// |A|(16x64) x ones(64x16) via 16 chained V_WMMA_F32_16X16X4_F32.
__global__ void __launch_bounds__(32) k_correct(
    const float* __restrict__ ysoft, const float* __restrict__ ytrue,
    const unsigned char* __restrict__ mask,
    float* yb /* in: smoothed_error, out: y (aliased on purpose, no restrict) */,
    float* __restrict__ last, const float* __restrict__ scal) {
  __shared__ float rs[16];
  int lane = threadIdx.x;
  int base = blockIdx.x * 16;      // NN % 16 == 0, so no partial blocks
  int row  = lane & 15;
  int half = lane >> 4;            // A layout: lanes 0-15 hold K pair {0,1}, 16-31 hold {2,3}

  const float* ap = yb + (size_t)(base + row) * CC + 2 * half;
  v2f b; b.x = 1.0f; b.y = 1.0f;   // all-ones B (layout-agnostic)
  v8f acc = {};
#pragma unroll
  for (int k = 0; k < 16; ++k) {   // channels 4k .. 4k+3
    v2f a;
    a.x = fabsf(ap[4 * k + 0]);
    a.y = fabsf(ap[4 * k + 1]);
    acc = __builtin_amdgcn_wmma_f32_16x16x4_f32(
        /*neg_a=*/false, a, /*neg_b=*/false, b,
        /*c_mod=*/(short)0, acc, /*reuse_a=*/false, /*reuse_b=*/false);
  }
  // D layout: vgpr r, lanes 0-15 -> M=r ; lanes 16-31 -> M=r+8 (all cols identical)
  if ((lane & 15) == 0) {
#pragma unroll
    for (int r = 0; r < 8; ++r) rs[r + 8 * half] = acc[r];
  }
  __syncthreads();

  float sigma = scal[0] / scal[1];  // sum|error| / numel
  for (int jj = 0; jj < 32; ++jj) { // 16 nodes * 64 ch / 32 lanes
    int t  = jj * 32 + lane;
    int il = t >> 6;
    int c  = t & 63;
    int g  = (base + il) * CC + c;
    float scale = sigma / rs[il];                      // 0-div -> inf, 0/0 -> nan
    if (__builtin_isinf(scale) || scale > 1000.0f) scale = 1.0f;
    float res = ysoft[g] + scale * yb[g];
    if (__builtin_isnan(res)) res = ysoft[g];
    float yv = mask[base + il] ? ytrue[g] : res;
    yb[g]   = yv;
    last[g] = (1.0f - A_SMOOTH) * yv;
  }
}

extern "C" void kernel_launch(void* const* d_in, const int* in_sizes, int n_in,
                              void* d_out, int out_size, void* d_ws, size_t ws_size,
                              hipStream_t stream) {
  const float* y_soft        = (const float*)d_in[0];
  const float* y_true        = (const float*)d_in[1];
  const int* src             = (const int*)d_in[2];
  const int* dst             = (const int*)d_in[3];
  const unsigned char* mask  = (const unsigned char*)d_in[4];
  float* out = (float*)d_out;

  float* ws       = (float*)d_ws;
  float* norm     = ws;
  float* buf_y    = ws + ((NN + 255) & ~255);   // 16B-aligned for float4
  float* buf_h    = buf_y + NCF;
  float* buf_last = buf_h + NCF;
  float* scal     = buf_last + NCF;             // [0]=sum|err| [1]=numel

  dim3 blk(256);
  int g_nc   = (NCF + 255) / 256;
  int g_e    = (EE + 255) / 256;
  int g_n    = (NN + 255) / 256;
  int g_edge = (EE * 16 + 255) / 256;

  k_init<<<g_nc, blk, 0, stream>>>(norm, buf_h, scal);
  k_deg<<<g_e, blk, 0, stream>>>(dst, norm);
  k_norm_fin<<<g_n, blk, 0, stream>>>(norm);
  k_error<<<g_nc, blk, 0, stream>>>(y_soft, y_true, mask, buf_y, buf_last, scal);

  for (int l = 0; l < 10; ++l) {
    k_edge<<<g_edge, blk, 0, stream>>>(src, dst, norm, buf_y, buf_h);
    k_combine<<<g_nc, blk, 0, stream>>>(norm, buf_last, buf_h, buf_y,
                                        A_CORR, -1.0f, 1.0f);
  }

  k_correct<<<NN / 16, dim3(32), 0, stream>>>(y_soft, y_true, mask,
                                              buf_y, buf_last, scal);

  for (int l = 0; l < 10; ++l) {
    k_edge<<<g_edge, blk, 0, stream>>>(src, dst, norm, buf_y, buf_h);
    float* yo = (l == 9) ? out : buf_y;
    k_combine<<<g_nc, blk, 0, stream>>>(norm, buf_last, buf_h, yo,
                                        A_SMOOTH, 0.0f, 1.0f);
  }
}